// Distribution_loss_5961414607235
// MI455X (gfx1250) — compile-verified
//
#include <hip/hip_runtime.h>

// ---------------------------------------------------------------------------
// KL distribution loss, MI455X (gfx1250, wave32).
// Memory-bound: 262 MB streamed once -> ~11us floor at 23.3 TB/s.
// Data path: global -> LDS via CDNA5 async copies (ASYNCcnt), LDS -> VGPR via
// ds_load_b128, wave32 butterfly reductions, f32 global atomics for the
// 1000-class segment sums, then a single-block finalize.
// ---------------------------------------------------------------------------

#define C_DIM        1000
#define ROW_BYTES    4000           // 1000 * 4B
#define ROW_PAD      4096           // padded LDS row (16B aligned, pow2)
#define WAVES_PER_BLOCK 4
#define THREADS      (WAVES_PER_BLOCK * 32)

__device__ __forceinline__ float wave_max(float v) {
#pragma unroll
    for (int off = 16; off >= 1; off >>= 1)
        v = fmaxf(v, __shfl_xor(v, off, 32));
    return v;
}

__device__ __forceinline__ float wave_sum(float v) {
#pragma unroll
    for (int off = 16; off >= 1; off >>= 1)
        v += __shfl_xor(v, off, 32);
    return v;
}

__global__ void zero_ws_kernel(float* __restrict__ p, int n) {
    int i = blockIdx.x * blockDim.x + threadIdx.x;
    if (i < n) p[i] = 0.0f;
}

__global__ __launch_bounds__(THREADS)
void kl_rows_kernel(const float* __restrict__ f1,
                    const float* __restrict__ f2,
                    const int*   __restrict__ label,
                    float* __restrict__ seg,
                    float* __restrict__ cnt,
                    int nrows)
{
    // 4 waves * 2 rows * 4096B = 32 KB static LDS
    __shared__ __align__(16) float smem[WAVES_PER_BLOCK * 2 * (ROW_PAD / 4)];

    const int tid  = threadIdx.x;
    const int wave = tid >> 5;
    const int lane = tid & 31;
    const int row  = blockIdx.x * WAVES_PER_BLOCK + wave;
    if (row >= nrows) return;   // wave-uniform

    // LDS byte addresses of this wave's staging area (flat shared ptr low
    // 32 bits == offset within the workgroup LDS allocation on AMDGCN).
    const unsigned lds_base = (unsigned)(size_t)(&smem[0]);
    const unsigned off1 = lds_base + (unsigned)(wave * 2 * ROW_PAD);
    const unsigned off2 = off1 + ROW_PAD;

    const char* g1 = (const char*)f1 + (size_t)row * ROW_BYTES;
    const char* g2 = (const char*)f2 + (size_t)row * ROW_BYTES;

    // ---- Async copy: 16 x global_load_async_to_lds_b128 (512B each) -------
    // Tail (i==7, lanes >= 26) is clamped to offset 0: duplicate 16B write of
    // identical data, benign; masked out of the math below.
#pragma unroll
    for (int i = 0; i < 8; ++i) {
        unsigned o  = (unsigned)(i * 512 + lane * 16);
        unsigned oc = (o < (unsigned)ROW_BYTES) ? o : 0u;
        unsigned long long a1 = (unsigned long long)(g1 + oc);
        unsigned long long a2 = (unsigned long long)(g2 + oc);
        unsigned l1 = off1 + oc;
        unsigned l2 = off2 + oc;
        asm volatile("global_load_async_to_lds_b128 %0, %1, off"
                     :: "v"(l1), "v"(a1) : "memory");
        asm volatile("global_load_async_to_lds_b128 %0, %1, off"
                     :: "v"(l2), "v"(a2) : "memory");
    }
    asm volatile("s_wait_asynccnt 0x0" ::: "memory");

    // ---- LDS -> VGPR (row lives in 64 VGPRs), ds_load_b128 x16 ------------
    const float4* b1 = (const float4*)(&smem[wave * 2 * (ROW_PAD / 4)]);
    const float4* b2 = (const float4*)(&smem[wave * 2 * (ROW_PAD / 4) + ROW_PAD / 4]);

    float4 a[8], b[8];
#pragma unroll
    for (int i = 0; i < 8; ++i) {
        a[i] = b1[i * 32 + lane];
        b[i] = b2[i * 32 + lane];
    }

    // ---- Pass 1: row maxes (numerical stabilization, matches reference) ---
    float m1 = -__builtin_inff(), m2 = -__builtin_inff();
#pragma unroll
    for (int i = 0; i < 8; ++i) {
        const float av[4] = { a[i].x, a[i].y, a[i].z, a[i].w };
        const float bv[4] = { b[i].x, b[i].y, b[i].z, b[i].w };
#pragma unroll
        for (int j = 0; j < 4; ++j) {
            int c = i * 128 + lane * 4 + j;
            if (c < C_DIM) {
                m1 = fmaxf(m1, av[j]);
                m2 = fmaxf(m2, bv[j]);
            }
        }
    }
    m1 = wave_max(m1);
    m2 = wave_max(m2);

    // ---- Pass 2 (fused): s1 = sum e^(f1-m1), s2 = sum e^(f2-m2),
    //      w = sum e^(f2-m2)*(f2-f1).
    // row_kl = (lse1 - lse2) + w / s2   (since sum t = 1)
    float s1 = 0.0f, s2 = 0.0f, w = 0.0f;
#pragma unroll
    for (int i = 0; i < 8; ++i) {
        const float av[4] = { a[i].x, a[i].y, a[i].z, a[i].w };
        const float bv[4] = { b[i].x, b[i].y, b[i].z, b[i].w };
#pragma unroll
        for (int j = 0; j < 4; ++j) {
            int c = i * 128 + lane * 4 + j;
            if (c < C_DIM) {
                float e1 = __expf(av[j] - m1);
                float e2 = __expf(bv[j] - m2);
                s1 += e1;
                s2 += e2;
                w  += e2 * (bv[j] - av[j]);
            }
        }
    }
    s1 = wave_sum(s1);
    s2 = wave_sum(s2);
    w  = wave_sum(w);

    if (lane == 0) {
        float lse1 = m1 + __logf(s1);
        float lse2 = m2 + __logf(s2);
        float kl   = (lse1 - lse2) + w / s2;
        int lab = label[row];
        atomicAdd(&seg[lab], kl);
        atomicAdd(&cnt[lab], 1.0f);
    }
}

__global__ __launch_bounds__(1024)
void kl_finalize_kernel(const float* __restrict__ seg,
                        const float* __restrict__ cnt,
                        float* __restrict__ out)
{
    __shared__ float red[32];
    const int tid = threadIdx.x;

    float v = 0.0f;
    if (tid < C_DIM) {
        float c = cnt[tid];
        v = (c > 0.0f) ? seg[tid] / (c * (float)C_DIM) : 0.0f;
    }
    v = wave_sum(v);
    if ((tid & 31) == 0) red[tid >> 5] = v;
    __syncthreads();
    if (tid < 32) {
        float x = red[tid];
        x = wave_sum(x);
        if (tid == 0) out[0] = x;
    }
}

extern "C" void kernel_launch(void* const* d_in, const int* in_sizes, int n_in,
                              void* d_out, int out_size, void* d_ws, size_t ws_size,
                              hipStream_t stream) {
    const float* f1    = (const float*)d_in[0];
    const float* f2    = (const float*)d_in[1];
    const int*   label = (const int*)d_in[2];

    float* seg = (float*)d_ws;          // [1024] (1000 used)
    float* cnt = seg + 1024;            // [1024]
    const int nrows = in_sizes[2];      // 32768

    // d_ws is poisoned; zero accumulators every call (deterministic).
    zero_ws_kernel<<<(2048 + 255) / 256, 256, 0, stream>>>(seg, 2048);

    const int blocks = (nrows + WAVES_PER_BLOCK - 1) / WAVES_PER_BLOCK;
    kl_rows_kernel<<<blocks, THREADS, 0, stream>>>(f1, f2, label, seg, cnt, nrows);

    kl_finalize_kernel<<<1, 1024, 0, stream>>>(seg, cnt, (float*)d_out);
}